// TopKSimilarActions_37563783971110
// MI455X (gfx1250) — compile-verified
//
#include <hip/hip_runtime.h>
#include <hip/hip_bf16.h>

typedef __attribute__((ext_vector_type(2))) float v2f;
typedef __attribute__((ext_vector_type(8))) float v8f;

#define NBATCH 4096
#define NACT   100000
#define DIM    512
#define TOPK   50

#define SPLIT  2             // action-space split across blocks
#define ACT_PER_SPLIT (NACT / SPLIT)
#define ROWS_PER_BLOCK 32
#define CHUNK   128          // actions per chunk (32 per wave = 2 N-tiles)
#define SSTRIDE 130          // scores staging row stride (floats): 8*130 % 64 == 16
#define ASTRIDE 516          // A-tile LDS row stride (floats): conflict-free, rows 16B aligned
#define NEG_INF (-3.402823466e38f)

__global__ __launch_bounds__(128, 2)
void topk_sim_kernel(const float* __restrict__ batch,
                     const float* __restrict__ actions,
                     float* __restrict__ ws_v,
                     int* __restrict__ ws_i)
{
    __shared__ float s_A[ROWS_PER_BLOCK * ASTRIDE];       // 66 KB resident A tile
    __shared__ float s_scores[ROWS_PER_BLOCK * SSTRIDE];  // 16.6 KB chunk staging
    __shared__ float s_vals[ROWS_PER_BLOCK][TOPK];
    __shared__ int   s_idxs[ROWS_PER_BLOCK][TOPK];
    __shared__ float s_min[ROWS_PER_BLOCK];
    __shared__ int   s_minpos[ROWS_PER_BLOCK];

    const int tid   = threadIdx.x;
    const int wave  = tid >> 5;
    const int lane  = tid & 31;
    const int l15   = lane & 15;
    const int khalf = lane >> 4;                       // 0: K=0,1   1: K=2,3
    const int rowBase = blockIdx.x * ROWS_PER_BLOCK;
    const int actBase = blockIdx.y * ACT_PER_SPLIT;
    const int actEnd  = actBase + ACT_PER_SPLIT;

    // ---- init per-row top-K state ----
    for (int i = tid; i < ROWS_PER_BLOCK * TOPK; i += blockDim.x) {
        ((float*)s_vals)[i] = NEG_INF;
        ((int*)s_idxs)[i]   = 0;
    }
    if (tid < ROWS_PER_BLOCK) { s_min[tid] = NEG_INF; s_minpos[tid] = 0; }

    // ---- stage the A tile (32 rows x 512 f32) into LDS via async copies ----
    // thread -> (row = tid>>2, 16B segments seg = tid&3 + 4*s): 32 B128 copies/thread
    {
        const int arow = tid >> 2;                     // 0..31
        const float* gsrc = batch + (size_t)(rowBase + arow) * DIM;
#pragma unroll
        for (int s = 0; s < 32; ++s) {
            int seg = (tid & 3) + s * 4;               // 0..127 (128 x 16B = 2KB row)
            const float* g = gsrc + seg * 4;
            unsigned l = (unsigned)(uintptr_t)&s_A[arow * ASTRIDE + seg * 4];
            asm volatile("global_load_async_to_lds_b128 %0, %1, off"
                         :: "v"(l), "v"(g) : "memory");
        }
        asm volatile("s_wait_asynccnt 0x0" ::: "memory");
    }
    __syncthreads();

    const int nChunks = (ACT_PER_SPLIT + CHUNK - 1) / CHUNK;
    for (int ch = 0; ch < nChunks; ++ch) {
        const int chunkStart = actBase + ch * CHUNK;

        // ======== phase 1: 32 rows x 128 actions via v_wmma_f32_16x16x4_f32 ========
        const float* bPtr[2];
#pragma unroll
        for (int t = 0; t < 2; ++t) {
            int n  = chunkStart + wave * 32 + t * 16 + l15;
            int nc = n < actEnd ? n : actEnd - 1;
            bPtr[t] = actions + (size_t)nc * DIM + khalf * 2;
            int np = n + CHUNK; np = np < actEnd ? np : actEnd - 1;
            __builtin_prefetch(actions + (size_t)np * DIM, 0, 1);
        }

        v8f acc00 = {}, acc01 = {}, acc10 = {}, acc11 = {};
        // K-loop in groups of 4 steps: 8 global B-loads + 8 LDS A-reads in flight,
        // then a 16-WMMA burst. Each B load feeds 2 row-tiles (0.5 loads/WMMA).
        for (int k = 0; k < DIM; k += 16) {
            v2f a0[4], a1[4], b0[4], b1[4];
#pragma unroll
            for (int s = 0; s < 4; ++s) {
                const int kk = k + 4 * s;
                a0[s] = *(const v2f*)&s_A[l15 * ASTRIDE + kk + khalf * 2];
                a1[s] = *(const v2f*)&s_A[(16 + l15) * ASTRIDE + kk + khalf * 2];
                b0[s] = *(const v2f*)(bPtr[0] + kk);
                b1[s] = *(const v2f*)(bPtr[1] + kk);
            }
#pragma unroll
            for (int s = 0; s < 4; ++s) {
                acc00 = __builtin_amdgcn_wmma_f32_16x16x4_f32(false, a0[s], false, b0[s], (short)0, acc00, false, false);
                acc01 = __builtin_amdgcn_wmma_f32_16x16x4_f32(false, a0[s], false, b1[s], (short)0, acc01, false, false);
                acc10 = __builtin_amdgcn_wmma_f32_16x16x4_f32(false, a1[s], false, b0[s], (short)0, acc10, false, false);
                acc11 = __builtin_amdgcn_wmma_f32_16x16x4_f32(false, a1[s], false, b1[s], (short)0, acc11, false, false);
            }
        }

        // stage: acc[rt][ct] VGPR r -> row rt*16 + r + 8*khalf, col wave*32 + ct*16 + l15
#pragma unroll
        for (int q = 0; q < 4; ++q) {
            const int rt = q >> 1, ct = q & 1;
            v8f acc = (q == 0) ? acc00 : (q == 1) ? acc01 : (q == 2) ? acc10 : acc11;
            int ncol = wave * 32 + ct * 16 + l15;
#pragma unroll
            for (int r = 0; r < 8; ++r) {
                s_scores[(rt * 16 + r + 8 * khalf) * SSTRIDE + ncol] = acc[r];
            }
        }
        __syncthreads();

        // ======== phase 2: each wave merges its 8 owned rows into the top-K ========
#pragma unroll
        for (int rr = 0; rr < 8; ++rr) {
            const int row = wave * 8 + rr;
            float curmin = s_min[row];
            int   minpos = s_minpos[row];
            for (int c0 = 0; c0 < CHUNK; c0 += 32) {
                int   c    = c0 + lane;
                int   gidx = chunkStart + c;
                float sc   = s_scores[row * SSTRIDE + c];
                bool  pred = (gidx < actEnd) && (sc > curmin);
                unsigned m = (unsigned)__ballot(pred);
                while (m) {
                    int src = __builtin_ctz(m);
                    m &= m - 1;
                    float cs = __shfl(sc, src, 32);
                    int   ci = __shfl(gidx, src, 32);
                    if (cs > curmin) {
                        if (lane == 0) {
                            s_vals[row][minpos] = cs;
                            s_idxs[row][minpos] = ci;
                        }
                        __builtin_amdgcn_wave_barrier();
                        // 32-lane butterfly arg-min over the 50 entries
                        float v0 = s_vals[row][lane];
                        int   p0 = lane;
                        float v1 = (lane + 32 < TOPK) ? s_vals[row][lane + 32] : 3.402823466e38f;
                        if (v1 < v0) { v0 = v1; p0 = lane + 32; }
#pragma unroll
                        for (int off = 16; off > 0; off >>= 1) {
                            float ov = __shfl_xor(v0, off, 32);
                            int   op = __shfl_xor(p0, off, 32);
                            if (ov < v0) { v0 = ov; p0 = op; }
                        }
                        curmin = v0;
                        minpos = p0;
                    }
                }
            }
            if (lane == 0) { s_min[row] = curmin; s_minpos[row] = minpos; }
        }
        __syncthreads();
    }

    // ======== emit this block's 50 candidates/row (rank-ordered) to workspace ========
#pragma unroll
    for (int rr = 0; rr < 8; ++rr) {
        const int row = wave * 8 + rr;
        const size_t base = ((size_t)(rowBase + row) * SPLIT + blockIdx.y) * TOPK;
        for (int e = lane; e < TOPK; e += 32) {
            float v  = s_vals[row][e];
            int   id = s_idxs[row][e];
            int rank = 0;
            for (int j = 0; j < TOPK; ++j) {
                float vj = s_vals[row][j];
                int   ij = s_idxs[row][j];
                rank += ((vj > v) || ((vj == v) && (ij < id))) ? 1 : 0;
            }
            ws_v[base + rank] = v;
            ws_i[base + rank] = id;
        }
    }
}

// Merge SPLIT*TOPK candidates per row -> final TOPK indices (desc, ties -> lower idx).
__global__ __launch_bounds__(128)
void topk_merge_kernel(const float* __restrict__ ws_v,
                       const int* __restrict__ ws_i,
                       int* __restrict__ out)
{
    const int row  = blockIdx.x * 4 + (threadIdx.x >> 5);
    const int lane = threadIdx.x & 31;
    const float* v = ws_v + (size_t)row * (SPLIT * TOPK);
    const int*   I = ws_i + (size_t)row * (SPLIT * TOPK);
    for (int e = lane; e < SPLIT * TOPK; e += 32) {
        float ve = v[e];
        int   ie = I[e];
        int rank = 0;
        for (int j = 0; j < SPLIT * TOPK; ++j) {
            float vj = v[j];
            int   ij = I[j];
            rank += ((vj > ve) || ((vj == ve) && (ij < ie))) ? 1 : 0;
        }
        if (rank < TOPK) out[(size_t)row * TOPK + rank] = ie;
    }
}

extern "C" void kernel_launch(void* const* d_in, const int* in_sizes, int n_in,
                              void* d_out, int out_size, void* d_ws, size_t ws_size,
                              hipStream_t stream) {
    const float* batch   = (const float*)d_in[0];   // [4096, 512]
    const float* actions = (const float*)d_in[1];   // [100000, 512]
    int* out = (int*)d_out;                         // [4096, 50] int32 indices

    // workspace: [4096][SPLIT][TOPK] float vals + [4096][SPLIT][TOPK] int idxs (3.3 MB)
    float* ws_v = (float*)d_ws;
    int*   ws_i = (int*)((char*)d_ws + (size_t)NBATCH * SPLIT * TOPK * sizeof(float));

    dim3 grid(NBATCH / ROWS_PER_BLOCK, SPLIT);      // 128 x 2 = 256 blocks
    dim3 block(128);                                // 4 wave32s
    hipLaunchKernelGGL(topk_sim_kernel, grid, block, 0, stream,
                       batch, actions, ws_v, ws_i);

    dim3 mgrid(NBATCH / 4);                         // 1024 blocks, 1 wave per row
    hipLaunchKernelGGL(topk_merge_kernel, mgrid, block, 0, stream,
                       ws_v, ws_i, out);
}